// aspect_rating_1_61538291417369
// MI455X (gfx1250) — compile-verified
//
#include <hip/hip_runtime.h>
#include <hip/hip_bf16.h>

// ---------------------------------------------------------------------------
// Problem constants (from the reference setup)
// ---------------------------------------------------------------------------
#define NREV   4096     // N reviews
#define LTOK   128      // tokens per review
#define DDIM   200      // word dimension
#define ADIM   32       // aspect dimension
#define NNEG   5
#define NB     1024     // users/items
#define FHIST  16384    // B*H flat history entries
#define AVG_RATING 3.5f

typedef __attribute__((ext_vector_type(2))) float v2f;
typedef __attribute__((ext_vector_type(8))) float v8f;

// ---------------------------------------------------------------------------
// Reduction helpers (deterministic: fixed shuffle tree + fixed serial fold)
// ---------------------------------------------------------------------------
__device__ __forceinline__ float wave_sum(float v) {
#pragma unroll
    for (int o = 16; o > 0; o >>= 1) v += __shfl_xor(v, o, 32);
    return v;
}
__device__ __forceinline__ float wave_max(float v) {
#pragma unroll
    for (int o = 16; o > 0; o >>= 1) v = fmaxf(v, __shfl_xor(v, o, 32));
    return v;
}

template <int NW>
__device__ __forceinline__ float block_sum(float v, float* red) {
    v = wave_sum(v);
    const int wid  = threadIdx.x >> 5;
    const int lane = threadIdx.x & 31;
    __syncthreads();                 // protect red[] against the previous call
    if (lane == 0) red[wid] = v;
    __syncthreads();
    float s = 0.f;
#pragma unroll
    for (int i = 0; i < NW; ++i) s += red[i];
    return s;                        // broadcast to every thread, fixed order
}

// ---------------------------------------------------------------------------
// Kernel 1: y = rev_pos @ M_w   [4096,200] x [200,200] -> [4096,200]
// fp32 WMMA 16x16x4. 256 row tiles x 13 col tiles (last col tile padded).
// One wave per 16x16 output tile, 8 waves per block. Column padding handled
// branchlessly (clamped index + 0/1 mask) so EXEC stays all-ones and the
// inner loop keeps clean b64 load clauses around each v_wmma.
// ---------------------------------------------------------------------------
__global__ void k1_y_gemm(const float* __restrict__ A,   // rev_pos [4096,200]
                          const float* __restrict__ Bm,  // M_w     [200,200]
                          float* __restrict__ Y) {       // out     [4096,200]
    const int wid   = threadIdx.x >> 5;
    const int lane  = threadIdx.x & 31;
    const int tile  = blockIdx.x * 8 + wid;     // exactly 256*13 tiles total
    const int ntile = 13;                        // ceil(200/16)
    const int mt = tile / ntile;
    const int nt = tile % ntile;

    const int mrow = mt * 16 + (lane & 15);      // A row for this lane
    const int ncol = nt * 16 + (lane & 15);      // B col for this lane
    const int koff = (lane >> 4) * 2;            // lane's K sub-offset
    const bool colok = (ncol < DDIM);
    const int  ncol_c = colok ? ncol : (DDIM - 1);   // clamped, loop-invariant
    const float cmask = colok ? 1.f : 0.f;           // zero out padded lanes

    const float* __restrict__ arow = A  + (long)mrow * DDIM + koff;
    const float* __restrict__ bcol = Bm + ncol_c;

    v8f c = {};
    for (int kb = 0; kb < DDIM; kb += 4) {
        v2f a, b;
        a.x = arow[kb];
        a.y = arow[kb + 1];
        b.x = bcol[(kb + koff) * DDIM] * cmask;
        b.y = bcol[(kb + koff + 1) * DDIM] * cmask;
        c = __builtin_amdgcn_wmma_f32_16x16x4_f32(false, a, false, b,
                                                  (short)0, c, false, false);
    }
    if (colok) {
#pragma unroll
        for (int v = 0; v < 8; ++v) {
            const int row = mt * 16 + v + 8 * (lane >> 4);
            Y[row * DDIM + ncol] = c[v];
        }
    }
}

// ---------------------------------------------------------------------------
// Kernel 2: fused per-review ABAE forward. One block (256 thr, 8 waves) per
// review. e_w tile (100 KB) lives in dynamic LDS (320 KB/WGP budget).
// Produces r_s[n,:] (global) and the per-review margin-loss partial.
// ---------------------------------------------------------------------------
__global__ void k2_review(const int*   __restrict__ hist,   // [N,L]
                          const float* __restrict__ Y,      // [N,200] from k1
                          const float* __restrict__ wemb,   // [VOCAB,200]
                          const float* __restrict__ Ww,     // [32,200]
                          const float* __restrict__ Wb,     // [32]
                          const float* __restrict__ Tw,     // [200,32]
                          const float* __restrict__ negs,   // [N*5,200]
                          float* __restrict__ Rs,           // out [N,200]
                          float* __restrict__ Jpart) {      // out [N]
    extern __shared__ float lds[];
    float* eW  = lds;            // 25600  flat e_w tile (row-major [L,D])
    float* yv  = eW  + LTOK*DDIM;// 200
    float* ax  = yv  + DDIM;     // 128    logits -> softmax weights
    float* zs  = ax  + LTOK;     // 200
    float* rs  = zs  + DDIM;     // 200
    float* pt  = rs  + DDIM;     // 32
    float* red = pt  + ADIM;     // 8

    const int n    = blockIdx.x;
    const int tid  = threadIdx.x;
    const int wid  = tid >> 5;
    const int lane = tid & 31;

    if (tid < DDIM) yv[tid] = Y[n * DDIM + tid];
    __syncthreads();

    // gather e_w into LDS + fused attention logits dx[l] = e_w[l,:].y
    for (int l = wid * 16; l < wid * 16 + 16; ++l) {
        const int tok = hist[n * LTOK + l];
        const float* src = wemb + (long)tok * DDIM;
        float acc = 0.f;
        for (int d = lane; d < DDIM; d += 32) {
            const float v = src[d];
            eW[l * DDIM + d] = v;
            acc += v * yv[d];
        }
        acc = wave_sum(acc);
        if (lane == 0) ax[l] = acc;
    }
    __syncthreads();

    // softmax over the 128 logits (wave 0)
    if (wid == 0) {
        float m = -3.4e38f;
        for (int i = lane; i < LTOK; i += 32) m = fmaxf(m, ax[i]);
        m = wave_max(m);
        float s = 0.f;
        for (int i = lane; i < LTOK; i += 32) {
            const float e = __expf(ax[i] - m);
            ax[i] = e;
            s += e;
        }
        s = wave_sum(s);
        const float inv = 1.f / s;
        for (int i = lane; i < LTOK; i += 32) ax[i] *= inv;
    }
    __syncthreads();

    // z_s via the reference's flat reshape: z_s[d] = sum_l flat[d*128+l]*ax[l]
    if (tid < DDIM) {
        float zd = 0.f;
        for (int i = 0; i < LTOK; ++i) {
            const int l = (i + tid) & (LTOK - 1);   // stagger: LDS-bank friendly
            zd += eW[tid * LTOK + l] * ax[l];
        }
        zs[tid] = zd;
    }
    __syncthreads();

    // p_t = softmax(z_s @ Ww^T + Wb) over A=32 (wave 0, lane = aspect)
    if (wid == 0) {
        float lg = Wb[lane];
        for (int d = 0; d < DDIM; ++d) lg += zs[d] * Ww[lane * DDIM + d];
        const float m = wave_max(lg);
        const float e = __expf(lg - m);
        const float s = wave_sum(e);
        pt[lane] = e / s;
    }
    __syncthreads();

    // r_s[d] = sum_a p_t[a] * T_w[d,a] ; store to global for rating head
    if (tid < DDIM) {
        float acc = 0.f;
#pragma unroll
        for (int a = 0; a < ADIM; ++a) acc += pt[a] * Tw[tid * ADIM + a];
        rs[tid] = acc;
        Rs[n * DDIM + tid] = acc;
    }

    const float rv = (tid < DDIM) ? rs[tid] : 0.f;
    const float zv = (tid < DDIM) ? zs[tid] : 0.f;
    const float rr = block_sum<8>(rv * rv, red);
    const float zz = block_sum<8>(zv * zv, red);
    const float rz = block_sum<8>(rv * zv, red);
    const float rn = fmaxf(sqrtf(rr), 1e-12f);
    const float zn = fmaxf(sqrtf(zz), 1e-12f);
    const float c1 = rz / (rn * zn);

    // 5 negatives: c2 = cos(z_n, r_s); accumulate relu(c2 - c1)
    float jl = 0.f;
#pragma unroll
    for (int j = 0; j < NNEG; ++j) {
        const float nv = (tid < DDIM)
            ? negs[((long)n * NNEG + j) * DDIM + tid] : 0.f;
        const float nn = block_sum<8>(nv * nv, red);
        const float nr = block_sum<8>(nv * rv, red);
        const float c2 = nr / (fmaxf(sqrtf(nn), 1e-12f) * rn);
        jl += fmaxf(0.f, c2 - c1);
    }
    if (tid == 0) Jpart[n] = jl;
}

// ---------------------------------------------------------------------------
// Kernel 3: orthogonality loss. Column-normalize T_w, then G = Tn^T Tn via
// four 16x16 f32 WMMA tiles (K=200), U = mean((G - I)^2). 1 block, 4 waves.
// ---------------------------------------------------------------------------
__global__ void k3_uloss(const float* __restrict__ Tw,   // [200,32]
                         float* __restrict__ Uout) {
    __shared__ float Tn[DDIM * ADIM];   // 25600 B
    __shared__ float inv[ADIM];
    __shared__ float red[4];

    const int tid  = threadIdx.x;
    const int wid  = tid >> 5;
    const int lane = tid & 31;

    if (tid < ADIM) {
        float s = 0.f;
        for (int d = 0; d < DDIM; ++d) {
            const float v = Tw[d * ADIM + tid];
            s += v * v;
        }
        inv[tid] = 1.f / fmaxf(sqrtf(s), 1e-12f);
    }
    __syncthreads();
    for (int k = tid; k < DDIM * ADIM; k += 128)
        Tn[k] = Tw[k] * inv[k & (ADIM - 1)];
    __syncthreads();

    // wave w -> G tile (rows 16*ti.., cols 16*tj..)
    const int ti = wid >> 1, tj = wid & 1;
    const int koff = (lane >> 4) * 2;
    const int iA = ti * 16 + (lane & 15);
    const int jB = tj * 16 + (lane & 15);

    v8f c = {};
    for (int kb = 0; kb < DDIM; kb += 4) {
        v2f a, b;
        a.x = Tn[(kb + koff) * ADIM + iA];      // A = Tn^T : A[i][k]=Tn[k][i]
        a.y = Tn[(kb + koff + 1) * ADIM + iA];
        b.x = Tn[(kb + koff) * ADIM + jB];      // B = Tn
        b.y = Tn[(kb + koff + 1) * ADIM + jB];
        c = __builtin_amdgcn_wmma_f32_16x16x4_f32(false, a, false, b,
                                                  (short)0, c, false, false);
    }
    float acc = 0.f;
#pragma unroll
    for (int v = 0; v < 8; ++v) {
        const int gi = ti * 16 + v + 8 * (lane >> 4);
        const int gj = tj * 16 + (lane & 15);
        const float e = c[v] - ((gi == gj) ? 1.f : 0.f);
        acc += e * e;
    }
    const float tot = block_sum<4>(acc, red);
    if (tid == 0) *Uout = tot / (float)(ADIM * ADIM);
}

// ---------------------------------------------------------------------------
// Kernel 4: rating head. seg_ids are sorted (repeat pattern) -> binary-search
// each segment's range, mean-pool r_s rows, dot(u,i)+3.5, squared error.
// One block per (user,item) pair b. Deterministic (no atomics).
// ---------------------------------------------------------------------------
__device__ __forceinline__ int lbound(const int* a, int n, int v) {
    int lo = 0, hi = n;
    while (lo < hi) { const int mid = (lo + hi) >> 1;
                      if (a[mid] < v) lo = mid + 1; else hi = mid; }
    return lo;
}

__global__ void k4_rating(const float* __restrict__ Rs,     // [N,200]
                          const int*   __restrict__ uidx,   // [F]
                          const int*   __restrict__ useg,   // [F] sorted
                          const int*   __restrict__ iidx,   // [F]
                          const int*   __restrict__ iseg,   // [F] sorted
                          const float* __restrict__ label,  // [B]
                          float* __restrict__ Rpart) {      // out [B]
    __shared__ float red[8];
    const int b   = blockIdx.x;
    const int tid = threadIdx.x;

    const int ulo = lbound(useg, FHIST, b), uhi = lbound(useg, FHIST, b + 1);
    const int ilo = lbound(iseg, FHIST, b), ihi = lbound(iseg, FHIST, b + 1);

    float ue = 0.f, ie = 0.f;
    if (tid < DDIM) {
        for (int k = ulo; k < uhi; ++k) ue += Rs[(long)uidx[k] * DDIM + tid];
        for (int k = ilo; k < ihi; ++k) ie += Rs[(long)iidx[k] * DDIM + tid];
        ue /= (float)(uhi - ulo);
        ie /= (float)(ihi - ilo);
    }
    const float dot = block_sum<8>(ue * ie, red);
    if (tid == 0) {
        const float diff = dot + AVG_RATING - label[b];
        Rpart[b] = diff * diff;
    }
}

// ---------------------------------------------------------------------------
// Kernel 5: finale. Fold J partials (N) and rating partials (B), combine with
// U loss -> (obj, rating_loss, abae_loss). Single block, fixed-order sums.
// ---------------------------------------------------------------------------
__global__ void k5_finish(const float* __restrict__ Jpart,
                          const float* __restrict__ Rpart,
                          const float* __restrict__ U,
                          float* __restrict__ out) {
    __shared__ float red[8];
    const int tid = threadIdx.x;
    float js = 0.f, rsum = 0.f;
    for (int i = tid; i < NREV; i += 256) js += Jpart[i];
    for (int i = tid; i < NB;   i += 256) rsum += Rpart[i];
    const float J = block_sum<8>(js, red)   / (float)(NREV * NNEG);
    const float R = block_sum<8>(rsum, red) / (float)NB;
    if (tid == 0) {
        const float abae = *U + J;
        out[0] = R + abae;   // obj
        out[1] = R;          // rating_loss
        out[2] = abae;       // abae_loss
    }
}

// ---------------------------------------------------------------------------
// Launch
// ---------------------------------------------------------------------------
extern "C" void kernel_launch(void* const* d_in, const int* in_sizes, int n_in,
                              void* d_out, int out_size, void* d_ws, size_t ws_size,
                              hipStream_t stream) {
    const int*   hist  = (const int*)  d_in[0];
    const float* rpos  = (const float*)d_in[1];
    const float* rneg  = (const float*)d_in[2];
    // d_in[3] user, d_in[4] item: unused by the reference
    const float* label = (const float*)d_in[5];
    const int*   uidx  = (const int*)  d_in[6];
    const int*   useg  = (const int*)  d_in[7];
    const int*   iidx  = (const int*)  d_in[8];
    const int*   iseg  = (const int*)  d_in[9];
    const float* wemb  = (const float*)d_in[10];
    const float* Mw    = (const float*)d_in[11];
    const float* Ww    = (const float*)d_in[12];
    const float* Wb    = (const float*)d_in[13];
    const float* Tw    = (const float*)d_in[14];
    float* out = (float*)d_out;

    float* ws   = (float*)d_ws;
    float* Y    = ws;                       // N*200
    float* Rs   = Y   + (size_t)NREV*DDIM;  // N*200
    float* Jp   = Rs  + (size_t)NREV*DDIM;  // N
    float* Rp   = Jp  + NREV;               // B
    float* U    = Rp  + NB;                 // 1

    // k1: 256 m-tiles x 13 n-tiles = 3328 waves, 8 waves/block
    k1_y_gemm<<<416, 256, 0, stream>>>(rpos, Mw, Y);

    // k2: one block per review; dynamic LDS for the 100KB e_w tile
    const size_t shmem =
        (size_t)(LTOK*DDIM + DDIM + LTOK + DDIM + DDIM + ADIM + 8) * sizeof(float);
    k2_review<<<NREV, 256, shmem, stream>>>(hist, Y, wemb, Ww, Wb, Tw, rneg,
                                            Rs, Jp);

    // k3: orthogonality loss (independent of k2; stream-ordered anyway)
    k3_uloss<<<1, 128, 0, stream>>>(Tw, U);

    // k4: rating head
    k4_rating<<<NB, 256, 0, stream>>>(Rs, uidx, useg, iidx, iseg, label, Rp);

    // k5: fold everything into the 3 output scalars
    k5_finish<<<1, 256, 0, stream>>>(Jp, Rp, U, out);
}